// GCA_12970801234182
// MI455X (gfx1250) — compile-verified
//
#include <hip/hip_runtime.h>
#include <hip/hip_bf16.h>
#include <math.h>

// Problem constants (from the reference)
#define BATCH 4
#define LSEQ 512
#define WSEQ 256
#define EDIM 300
#define HDIM 768
#define NLAB 6

typedef float v2f __attribute__((ext_vector_type(2)));
typedef float v8f __attribute__((ext_vector_type(8)));

// ---------------------------------------------------------------------------
// Kernel 1: a[bw][h] = emb_a[word_seq[bw]] @ lin_w + lin_b
//   M = B*W = 1024, K = 300, N = 768.  One wave -> one 16x16 tile of `a`.
//   fp32 WMMA 16x16x4:
//     A (16x4):  lane L holds M = L%16, K-pair = 2*(L/16) .. +1  (2 VGPRs)
//     B (4x16):  lane L holds N = L%16, K-pair = 2*(L/16) .. +1  (2 VGPRs)
//     C (16x16): vgpr i, lane L -> M = i + 8*(L/16), N = L%16    (8 VGPRs)
// ---------------------------------------------------------------------------
__global__ __launch_bounds__(32)
void ga_proj_kernel(const int* __restrict__ word_seq,
                    const float* __restrict__ emb_a,
                    const float* __restrict__ lin_w,
                    const float* __restrict__ lin_b,
                    float* __restrict__ a_out) {
    const int lane   = threadIdx.x;          // 0..31 (wave32)
    const int n_tile = blockIdx.x;           // 0..47  (768/16)
    const int m_tile = blockIdx.y;           // 0..63  (1024/16)

    const int m    = lane & 15;              // A row this lane feeds
    const int n    = lane & 15;              // B col this lane feeds
    const int koff = (lane >> 4) << 1;       // 0 or 2: which K pair

    const int m_glob = m_tile * 16 + m;      // flat [B*W) row
    const int n_base = n_tile * 16;

    const int   widx = word_seq[m_glob];
    const float* arow = emb_a + (size_t)widx * EDIM;

    v8f acc = {};
    for (int k = 0; k < EDIM; k += 4) {
        v2f av, bv;
        av.x = arow[k + koff];
        av.y = arow[k + koff + 1];
        bv.x = lin_w[(size_t)(k + koff)     * HDIM + n_base + n];
        bv.y = lin_w[(size_t)(k + koff + 1) * HDIM + n_base + n];
        acc = __builtin_amdgcn_wmma_f32_16x16x4_f32(
            false, av, false, bv, (short)0, acc, false, false);
    }

    const float bias  = lin_b[n_base + n];
    const int   mhalf = (lane >> 4) << 3;    // lanes 16-31 hold M = i+8
#pragma unroll
    for (int i = 0; i < 8; ++i) {
        const int m_out = m_tile * 16 + mhalf + i;
        a_out[(size_t)m_out * HDIM + n_base + n] = acc[i] + bias;
    }
}

// ---------------------------------------------------------------------------
// Kernel 2: u[b][l][w] = (hidden[b,l,:] . a[b,w,:]) / sqrt(H)
//   Per batch: M = 512 (l), N = 256 (w), K = 768 (h).
//   B operand B[k][n] = a[b, n, k]  (a rows read as columns).
// ---------------------------------------------------------------------------
__global__ __launch_bounds__(32)
void ga_score_kernel(const float* __restrict__ hidden,
                     const float* __restrict__ a_ws,
                     float* __restrict__ u_out) {
    const int lane   = threadIdx.x;
    const int w_tile = blockIdx.x;           // 0..15  (256/16)
    const int l_tile = blockIdx.y;           // 0..31  (512/16)
    const int b      = blockIdx.z;           // 0..3

    const int m    = lane & 15;
    const int n    = lane & 15;
    const int koff = (lane >> 4) << 1;

    const float* hrow = hidden + ((size_t)b * LSEQ + l_tile * 16 + m) * HDIM;
    const float* acol = a_ws   + ((size_t)b * WSEQ + w_tile * 16 + n) * HDIM;

    v8f acc = {};
    for (int k = 0; k < HDIM; k += 4) {
        v2f av, bv;
        av.x = hrow[k + koff];
        av.y = hrow[k + koff + 1];
        bv.x = acol[k + koff];               // B[k][n] = a[n][k]
        bv.y = acol[k + koff + 1];
        acc = __builtin_amdgcn_wmma_f32_16x16x4_f32(
            false, av, false, bv, (short)0, acc, false, false);
    }

    const float scale = 0.036084391824351615f;  // 1 / sqrt(768)
    const int   mhalf = (lane >> 4) << 3;
#pragma unroll
    for (int i = 0; i < 8; ++i) {
        const int l = l_tile * 16 + mhalf + i;
        u_out[((size_t)b * LSEQ + l) * WSEQ + w_tile * 16 + n] = acc[i] * scale;
    }
}

// ---------------------------------------------------------------------------
// Kernel 3: fused masked exp-normalize + label-bucket reduction + emb_c matvec
//   For each row (b,l):
//     e[w]  = exp(u[w]) * (label[w] != 0)
//     tot   = sum_w e[w]
//     s[k]  = (sum_{w: label==k} e[w]) / (tot + 1e-10)
//     o[h]  = sum_k s[k] * emb_c[k][h]
//   One wave per row; 8 u-elements per lane; butterfly shfl reductions.
// ---------------------------------------------------------------------------
__global__ __launch_bounds__(256)
void ga_out_kernel(const float* __restrict__ u_ws,
                   const int* __restrict__ labels,
                   const float* __restrict__ emb_c,
                   float* __restrict__ out) {
    const int lane = threadIdx.x;                       // 0..31
    const int row  = blockIdx.x * blockDim.y + threadIdx.y;  // 0..B*L-1

    const float* urow = u_ws   + (size_t)row * WSEQ;
    const int*   lrow = labels + (size_t)row * WSEQ;

    float s[NLAB];
#pragma unroll
    for (int k = 0; k < NLAB; ++k) s[k] = 0.0f;
    float tot = 0.0f;

#pragma unroll
    for (int j = 0; j < WSEQ / 32; ++j) {
        const int w  = j * 32 + lane;
        const int lv = lrow[w];
        const float mask = (lv > 0) ? 1.0f : 0.0f;
        const float e    = __expf(urow[w]) * mask;
        tot += e;
        s[lv] += e;   // lv==0 contributes e==0; harmless
    }

    // Butterfly reductions: all lanes end with the full sums.
#pragma unroll
    for (int off = 16; off > 0; off >>= 1) {
        tot += __shfl_xor(tot, off, 32);
#pragma unroll
        for (int k = 0; k < NLAB; ++k)
            s[k] += __shfl_xor(s[k], off, 32);
    }

    const float inv = 1.0f / (tot + 1e-10f);
#pragma unroll
    for (int k = 0; k < NLAB; ++k) s[k] *= inv;

    // o[row, h] = sum_k s[k] * emb_c[k][h]; 768/32 = 24 channels per lane.
    float* orow = out + (size_t)row * HDIM;
#pragma unroll
    for (int j = 0; j < HDIM / 32; ++j) {
        const int h = j * 32 + lane;
        float acc = 0.0f;
#pragma unroll
        for (int k = 0; k < NLAB; ++k)
            acc = fmaf(s[k], emb_c[k * HDIM + h], acc);
        orow[h] = acc;
    }
}

// ---------------------------------------------------------------------------
extern "C" void kernel_launch(void* const* d_in, const int* in_sizes, int n_in,
                              void* d_out, int out_size, void* d_ws, size_t ws_size,
                              hipStream_t stream) {
    const int*   word_seq = (const int*)d_in[0];    // [B, W]
    const float* hidden   = (const float*)d_in[1];  // [B, L, H]
    const int*   labels   = (const int*)d_in[2];    // [B, L, W]
    const float* emb_a    = (const float*)d_in[3];  // [VOCAB, E]
    const float* lin_w    = (const float*)d_in[4];  // [E, H]
    const float* lin_b    = (const float*)d_in[5];  // [H]
    const float* emb_c    = (const float*)d_in[6];  // [NLAB, H]
    float*       out      = (float*)d_out;          // [B, L, H]

    float* a_ws = (float*)d_ws;                       // [B*W, H]   = 3 MB
    float* u_ws = a_ws + (size_t)BATCH * WSEQ * HDIM; // [B*L, W]   = 2 MB

    // K1: gathered projection GEMM  (1024 x 768, K=300)
    ga_proj_kernel<<<dim3(HDIM / 16, (BATCH * WSEQ) / 16), dim3(32), 0, stream>>>(
        word_seq, emb_a, lin_w, lin_b, a_ws);

    // K2: attention scores  (per batch 512 x 256, K=768)
    ga_score_kernel<<<dim3(WSEQ / 16, LSEQ / 16, BATCH), dim3(32), 0, stream>>>(
        hidden, a_ws, u_ws);

    // K3: fused exp-normalize + 6-label bucketing + emb_c matvec
    ga_out_kernel<<<dim3((BATCH * LSEQ) / 8), dim3(32, 8), 0, stream>>>(
        u_ws, labels, emb_c, out);
}